// LocallyConnected2D__1125281431911
// MI455X (gfx1250) — compile-verified
//
#include <hip/hip_runtime.h>

typedef __attribute__((ext_vector_type(2))) float v2f;
typedef __attribute__((ext_vector_type(8))) float v8f;

// Problem geometry (fixed by the reference): B=64, H=W=16, Cin=Cout=32.
constexpr int BATCH = 64;
constexpr int HW    = 256;   // H*W spatial positions
constexpr int CIN   = 32;
constexpr int COUT  = 32;
constexpr int KROW  = HW * COUT;              // 8192: stride between ci rows in kernel tensor
constexpr int LDSX_STRIDE = 36;               // padded row stride (floats); 144 B = 16B-aligned rows
constexpr int LDSK_STRIDE = 36;

__global__ __launch_bounds__(256) void lc2d_diag_wmma_f32(
    const float* __restrict__ inp,   // (B, H, W, Cin)  = (b*256 + p)*32 + ci
    const float* __restrict__ ker,   // (H,W,Cin,H,W,Cout); diagonal block of p at
                                     //   p*(Cin*KROW + Cout) + ci*KROW + co
    const float* __restrict__ bias,  // (H, W, Cout) = p*32 + co
    float* __restrict__ out)         // (B, H, W, Cout) = (b*256 + p)*32 + co
{
    __shared__ __attribute__((aligned(16))) float ldsX[BATCH * LDSX_STRIDE]; // 64 x 32 padded
    __shared__ __attribute__((aligned(16))) float ldsK[CIN   * LDSK_STRIDE]; // 32 x 32 padded

    const int p = blockIdx.x;    // spatial position 0..255
    const int t = threadIdx.x;   // 0..255

    // ---- Stage K_p (diagonal 32x32 block) into LDS: 256 threads x float4
    {
        const long diagBase = (long)p * (CIN * KROW + COUT);
        const int ci = t >> 3;          // 0..31
        const int co = (t & 7) << 2;    // 0,4,...,28
        const float4 v = *(const float4*)(ker + diagBase + (long)ci * KROW + co);
        *(float4*)(&ldsK[ci * LDSK_STRIDE + co]) = v;
    }
    // ---- Stage X_p (64x32) into LDS: 256 threads x 2 float4
    {
#pragma unroll
        for (int c = 0; c < 2; ++c) {
            const int chunk = t + c * 256;    // 0..511
            const int row   = chunk >> 3;     // batch row 0..63
            const int seg   = (chunk & 7) << 2;
            const float4 v = *(const float4*)(inp + ((long)row * HW + p) * CIN + seg);
            *(float4*)(&ldsX[row * LDSX_STRIDE + seg]) = v;
        }
    }
    __syncthreads();

    const int lane = t & 31;
    const int wave = t >> 5;            // 0..7
    const int lo   = lane & 15;         // column / row-in-tile selector
    const int hi   = lane >> 4;         // half-wave selector
    const int m0   = (wave >> 1) << 4;  // batch-tile origin: 0,16,32,48
    const int n0   = (wave & 1) << 4;   // cout-tile origin: 0,16

    // C/D tile accumulator: VGPR v of lane holds D[m0 + v + 8*hi][n0 + lo]
    v8f acc = {};

#pragma unroll
    for (int s = 0; s < 8; ++s) {
        const int kA = (s << 2) + (hi << 1);  // K base for this lane's VGPR pair
        // A (16x4 fp32): lane row = m0+lo, VGPR v -> K = kA + v  (8B-aligned LDS load)
        const v2f a = *(const v2f*)(&ldsX[(m0 + lo) * LDSX_STRIDE + kA]);
        // B (4x16 fp32): VGPR v -> row K = kA + v, col = n0+lo
        v2f b;
        b.x = ldsK[kA * LDSK_STRIDE + n0 + lo];
        b.y = ldsK[(kA + 1) * LDSK_STRIDE + n0 + lo];
        acc = __builtin_amdgcn_wmma_f32_16x16x4_f32(
            /*neg_a=*/false, a, /*neg_b=*/false, b,
            /*c_mod=*/(short)0, acc, /*reuse_a=*/false, /*reuse_b=*/false);
    }

    // Bias depends only on the N column of the tile -> one load per lane.
    const float bv = bias[p * COUT + n0 + lo];

#pragma unroll
    for (int v = 0; v < 8; ++v) {
        const int bg = m0 + v + (hi << 3);    // global batch row of VGPR v
        out[((long)bg * HW + p) * COUT + n0 + lo] = acc[v] + bv;
    }
}

extern "C" void kernel_launch(void* const* d_in, const int* in_sizes, int n_in,
                              void* d_out, int out_size, void* d_ws, size_t ws_size,
                              hipStream_t stream) {
    const float* inp  = (const float*)d_in[0];  // inputs (64,16,16,32) fp32
    const float* ker  = (const float*)d_in[1];  // kernel (16,16,32,16,16,32) fp32
    const float* bias = (const float*)d_in[2];  // bias (16,16,32) fp32
    float* out = (float*)d_out;                 // (64,16,16,32) fp32

    lc2d_diag_wmma_f32<<<dim3(HW), dim3(256), 0, stream>>>(inp, ker, bias, out);
}